// Loss_45200235823826
// MI455X (gfx1250) — compile-verified
//
#include <hip/hip_runtime.h>
#include <hip/hip_bf16.h>

// ---- problem constants (from reference) ----
#define BATCH   16384
#define SGRID   7
#define CELLS   49          // 7*7
#define CH      30
#define TILE    (CELLS*CH)  // 1470 floats per image
#define MBOX    16
#define NCLS    20
#define IMGS    8           // images per block (one per wave32)
#define THREADS 256
#define NBLK    (BATCH/IMGS) // 2048

typedef __attribute__((ext_vector_type(2))) float v2f;
typedef __attribute__((ext_vector_type(8))) float v8f;

// async-to-LDS b128 element type: must match builtin's expected
// 'int __attribute__((vector_size(16)))*' parameter type exactly.
typedef int b128_t __attribute__((vector_size(16)));
typedef __attribute__((address_space(1))) b128_t gb128_t;
typedef __attribute__((address_space(3))) b128_t lb128_t;

__device__ __forceinline__ float iou_cf(float ax, float ay, float aw, float ah,
                                        float bx, float by, float bw, float bh) {
  float ax1 = ax - aw * 0.5f, ay1 = ay - ah * 0.5f;
  float ax2 = ax + aw * 0.5f, ay2 = ay + ah * 0.5f;
  float bx1 = bx - bw * 0.5f, by1 = by - bh * 0.5f;
  float bx2 = bx + bw * 0.5f, by2 = by + bh * 0.5f;
  float iw = fmaxf(fminf(ax2, bx2) - fmaxf(ax1, bx1), 0.0f);
  float ih = fmaxf(fminf(ay2, by2) - fmaxf(ay1, by1), 0.0f);
  float inter = iw * ih;
  float uni = aw * ah + bw * bh - inter;
  return inter / (uni + 1e-6f);
}

// Wave32 all-lane sum. Uses V_WMMA_F32_16X16X4_F32 with B = ones:
//   D[m][n] = sum_k A[m][k]  -> rowsum = p[m] + p[m+16] (A layout per ISA 7.12.2).
// Sum of D VGPRs 0..7 gives half-sums in lane-halves; one shfl_xor(16) finishes.
// EXEC must be all ones at the call site (kernel keeps top-level flow branch-free).
__device__ __forceinline__ float wave_sum32(float p) {
#if __has_builtin(__builtin_amdgcn_wmma_f32_16x16x4_f32)
  v2f a = {p, 0.0f};
  v2f ones = {1.0f, 1.0f};
  v8f c = {};
  v8f d = __builtin_amdgcn_wmma_f32_16x16x4_f32(false, a, false, ones,
                                                (short)0, c, false, false);
  float s = ((d[0] + d[1]) + (d[2] + d[3])) + ((d[4] + d[5]) + (d[6] + d[7]));
  s += __shfl_xor(s, 16, 32);
  return s;
#else
  for (int off = 16; off > 0; off >>= 1) p += __shfl_xor(p, off, 32);
  return p;
#endif
}

__global__ __launch_bounds__(THREADS) void yolo_loss_partial(
    const float* __restrict__ pred,
    const float* __restrict__ gt_xywh,
    const int* __restrict__ gt_class,
    float* __restrict__ partial) {
  __shared__ __align__(16) float smem[IMGS * TILE];  // 47040 B
  __shared__ float waveTot[IMGS];

  const int tid = threadIdx.x;
  const float* gbase = pred + (size_t)blockIdx.x * (IMGS * TILE);

  // ---- stage 8 image tiles into LDS: 11760 floats = 2940 x b128, streaming ----
#if __has_builtin(__builtin_amdgcn_global_load_async_to_lds_b128)
  for (int i = tid; i < (IMGS * TILE / 4); i += THREADS) {
    __builtin_amdgcn_global_load_async_to_lds_b128(
        (gb128_t*)(gbase + i * 4), (lb128_t*)(smem + i * 4), 0, 0);
  }
#if __has_builtin(__builtin_amdgcn_s_wait_asynccnt)
  __builtin_amdgcn_s_wait_asynccnt(0);
#else
  asm volatile("s_wait_asynccnt 0" ::: "memory");
#endif
#else
  for (int i = tid; i < (IMGS * TILE / 4); i += THREADS) {
    float4 v = ((const float4*)gbase)[i];
    ((float4*)smem)[i] = v;
  }
#endif
  __syncthreads();

  const int lane = tid & 31;
  const int wave = tid >> 5;
  const int b = blockIdx.x * IMGS + wave;
  const float* tile = smem + wave * TILE;

  // ---- per-box work: lanes 0-15 own boxes; lanes 16-31 mirror (masked out) ----
  const int m = lane & 15;
  float4 g = ((const float4*)gt_xywh)[b * MBOX + m];
  int cls = gt_class[b * MBOX + m];

  // faithful to reference: row from x, col from y
  int row = (int)floorf(g.x * (float)SGRID);
  row = row < 0 ? 0 : (row > SGRID - 1 ? SGRID - 1 : row);
  int col = (int)floorf(g.y * (float)SGRID);
  col = col < 0 ? 0 : (col > SGRID - 1 ? SGRID - 1 : col);
  int cell = row * SGRID + col;

  // dedup (first occurrence per cell) + object-cell mask, via 16 shuffles
  unsigned long long objmask = 0ull;
  bool keep = (lane < 16);
#pragma unroll
  for (int j = 0; j < 16; ++j) {
    int cj = __shfl(cell, j, 32);
    objmask |= (1ull << cj);
    keep = keep && !((j < m) && (cj == cell));
  }

  // gather this box's cell prediction from LDS (stride 30 dwords: conflict-free)
  float cp[CH];
#pragma unroll
  for (int c = 0; c < CH; ++c) cp[c] = tile[cell * CH + c];

  float tx = g.x * (float)SGRID - (float)col;
  float ty = g.y * (float)SGRID - (float)row;
  float tw = sqrtf(g.z);
  float th = sqrtf(g.w);

  float i1 = iou_cf(cp[0], cp[1], cp[2], cp[3], tx, ty, tw, th);
  float i2 = iou_cf(cp[5], cp[6], cp[7], cp[8], tx, ty, tw, th);
  bool use2 = (i2 >= i1);
  float sx = use2 ? cp[5] : cp[0];
  float sy = use2 ? cp[6] : cp[1];
  float sw = use2 ? cp[7] : cp[2];
  float sh = use2 ? cp[8] : cp[3];
  float sconf = use2 ? cp[9] : cp[4];
  float siou = use2 ? i2 : i1;

  float dx = sx - tx, dy = sy - ty, dw = sw - tw, dh = sh - th;
  float term = 5.0f * (dx * dx + dy * dy + dw * dw + dh * dh);  // LAMBDA_COORD
  float dc = sconf - siou;                                      // conf target detached
  term += dc * dc;
  float clt = 0.0f;
#pragma unroll
  for (int c = 0; c < NCLS; ++c) {
    float d = cp[10 + c] - ((c == cls) ? 1.0f : 0.0f);
    clt += d * d;
  }
  term += clt;
  float p = keep ? term : 0.0f;

  // ---- no-object term: each lane covers cells `lane` and `lane+32` ----
  {
    int c1 = lane;  // always < 49
    float a4 = tile[c1 * CH + 4], a9 = tile[c1 * CH + 9];
    bool o1 = (objmask >> c1) & 1ull;
    p += o1 ? 0.0f : 0.5f * (a4 * a4 + a9 * a9);  // LAMBDA_NOOBJ

    int c2 = lane + 32;
    bool v2 = (c2 < CELLS);
    int c2c = v2 ? c2 : 0;
    float b4 = tile[c2c * CH + 4], b9 = tile[c2c * CH + 9];
    bool o2 = (objmask >> c2c) & 1ull;
    p += (v2 && !o2) ? 0.5f * (b4 * b4 + b9 * b9) : 0.0f;
  }

  // ---- deterministic block partial: WMMA wave-sum, then wave 0 folds 8 waves ----
  float wsum = wave_sum32(p);
  if (lane == 0) waveTot[wave] = wsum;
  __syncthreads();
  if (tid == 0) {
    float t = 0.0f;
#pragma unroll
    for (int w = 0; w < IMGS; ++w) t += waveTot[w];
    partial[blockIdx.x] = t;
  }
}

__global__ __launch_bounds__(THREADS) void yolo_loss_reduce(
    const float* __restrict__ partial, float* __restrict__ out, int n,
    float scale) {
  __shared__ float sh[THREADS];
  float s = 0.0f;
  for (int i = threadIdx.x; i < n; i += THREADS) s += partial[i];
  sh[threadIdx.x] = s;
  __syncthreads();
  for (int off = THREADS / 2; off > 0; off >>= 1) {
    if (threadIdx.x < off) sh[threadIdx.x] += sh[threadIdx.x + off];
    __syncthreads();
  }
  if (threadIdx.x == 0) out[0] = sh[0] * scale;
}

extern "C" void kernel_launch(void* const* d_in, const int* in_sizes, int n_in,
                              void* d_out, int out_size, void* d_ws,
                              size_t ws_size, hipStream_t stream) {
  (void)in_sizes; (void)n_in; (void)out_size; (void)ws_size;
  const float* pred = (const float*)d_in[0];
  const float* gt_xywh = (const float*)d_in[1];
  const int* gt_class = (const int*)d_in[2];
  float* partial = (float*)d_ws;  // NBLK floats = 8 KB scratch

  yolo_loss_partial<<<NBLK, THREADS, 0, stream>>>(pred, gt_xywh, gt_class,
                                                  partial);
  yolo_loss_reduce<<<1, THREADS, 0, stream>>>(partial, (float*)d_out, NBLK,
                                              1.0f / (float)BATCH);
}